// GCNKnorm_40956808135033
// MI455X (gfx1250) — compile-verified
//
#include <hip/hip_runtime.h>
#include <math.h>

typedef __attribute__((ext_vector_type(2))) float v2f;
typedef __attribute__((ext_vector_type(8))) float v8f;

#define NNODES 50000
#define NEDGES 800000

// ---------------------------------------------------------------------------
// GEMM: H[n, NC] = X[n, 128] @ W[128, NC]   via V_WMMA_F32_16X16X4_F32
// one wave -> 16 output rows x all column tiles.
// W cached in LDS packed as k-pairs: lw[(k/2)*PC + c] = {W[k][c], W[k+1][c]}
// so each B fragment is ONE aligned ds_load_b64 straight into a VGPR pair.
// ---------------------------------------------------------------------------
template <int NC>
__global__ __launch_bounds__(256) void gemm_wmma_f32(
    const float* __restrict__ X, const float* __restrict__ W,
    float* __restrict__ H, int n) {
  constexpr int K  = 128;
  constexpr int CT = (NC + 15) / 16;   // column tiles
  constexpr int PC = CT * 16;          // padded columns
  __shared__ v2f lw[(K / 2) * PC];

  const int tid = threadIdx.x;
  for (int i = tid; i < (K / 2) * PC; i += 256) {
    const int kp = i / PC, c = i % PC;
    v2f val;
    val.x = (c < NC) ? W[(2 * kp) * NC + c] : 0.0f;
    val.y = (c < NC) ? W[(2 * kp + 1) * NC + c] : 0.0f;
    lw[i] = val;
  }
  __syncthreads();

  const int wave = tid >> 5;
  const int lane = tid & 31;
  const int row0 = (blockIdx.x * 8 + wave) * 16;
  if (row0 >= n) return;                 // wave-uniform; EXEC stays all-1s for WMMA

  const int m  = lane & 15;              // row (A) / col (B,C) within tile
  const int hk = lane >> 4;              // half-wave selector

  v8f acc[CT];
#pragma unroll
  for (int t = 0; t < CT; ++t) acc[t] = {};

  const float* xrow = X + (size_t)(row0 + m) * K + 2 * hk;

  for (int k4 = 0; k4 < K; k4 += 4) {
    // A fragment: vgpr0 = A[m][k4+2*hk], vgpr1 = A[m][k4+2*hk+1]
    v2f a;
    a.x = xrow[k4];
    a.y = xrow[k4 + 1];
    // B fragment rows (k4+2*hk, k4+2*hk+1) = packed pair index k4/2 + hk
    const v2f* brow = &lw[(k4 / 2 + hk) * PC + m];
#pragma unroll
    for (int t = 0; t < CT; ++t) {
      const v2f b = brow[t * 16];
      acc[t] = __builtin_amdgcn_wmma_f32_16x16x4_f32(
          false, a, false, b, (short)0, acc[t], false, false);
    }
  }

  // C/D layout: vgpr r -> row (r + 8*hk), col = tile*16 + m
  const int crow = row0 + hk * 8;
#pragma unroll
  for (int t = 0; t < CT; ++t) {
    const int col = t * 16 + m;
    if ((NC % 16 == 0) || (col < NC)) {
#pragma unroll
      for (int r = 0; r < 8; ++r)
        H[(size_t)(crow + r) * NC + col] = acc[t][r];
    }
  }
}

// ---------------------------------------------------------------------------
// Zero fill
// ---------------------------------------------------------------------------
__global__ void zerof(float* __restrict__ p, long long n) {
  long long i = (long long)blockIdx.x * blockDim.x + threadIdx.x;
  if (i < n) p[i] = 0.0f;
}

// ---------------------------------------------------------------------------
// Edge aggregation, 128 features: one wave per edge, float4 per lane
// A[tgt[e], :] += w[e] * H[src[e], :]
// ---------------------------------------------------------------------------
__global__ __launch_bounds__(256) void agg128(
    const float* __restrict__ H, const int* __restrict__ src,
    const int* __restrict__ tgt, const float* __restrict__ w,
    float* __restrict__ A, int E) {
  const int e = (int)(((long long)blockIdx.x * blockDim.x + threadIdx.x) >> 5);
  const int lane = threadIdx.x & 31;
  if (e >= E) return;
  const int   s  = src[e];
  const int   t  = tgt[e];
  const float we = w[e];
  const float4 v = *((const float4*)(H + (size_t)s * 128) + lane);
  float* ap = A + (size_t)t * 128 + lane * 4;
  atomicAdd(ap + 0, we * v.x);
  atomicAdd(ap + 1, we * v.y);
  atomicAdd(ap + 2, we * v.z);
  atomicAdd(ap + 3, we * v.w);
}

// Edge aggregation, 40 features: one wave per edge, lanes cover f and f+32
__global__ __launch_bounds__(256) void agg40(
    const float* __restrict__ H, const int* __restrict__ src,
    const int* __restrict__ tgt, const float* __restrict__ w,
    float* __restrict__ A, int E) {
  const int e = (int)(((long long)blockIdx.x * blockDim.x + threadIdx.x) >> 5);
  const int lane = threadIdx.x & 31;
  if (e >= E) return;
  const int   s  = src[e];
  const int   t  = tgt[e];
  const float we = w[e];
  atomicAdd(A + (size_t)t * 40 + lane, we * H[(size_t)s * 40 + lane]);
  if (lane < 8)
    atomicAdd(A + (size_t)t * 40 + 32 + lane, we * H[(size_t)s * 40 + 32 + lane]);
}

// ---------------------------------------------------------------------------
// Elementwise bias + ReLU (128 features)
// ---------------------------------------------------------------------------
__global__ void bias_relu128(float* __restrict__ A, const float* __restrict__ b,
                             long long total) {
  long long i = (long long)blockIdx.x * blockDim.x + threadIdx.x;
  if (i >= total) return;
  const int c = (int)(i & 127);
  A[i] = fmaxf(A[i] + b[c], 0.0f);
}

// ---------------------------------------------------------------------------
// bias + ReLU + GroupNorm(32 groups x 4 ch): one wave per node, lane = group
// (group stats entirely lane-local: 4 channels per group)
// ---------------------------------------------------------------------------
__global__ __launch_bounds__(256) void gn_bias_relu(
    float* __restrict__ A, const float* __restrict__ b,
    const float* __restrict__ gamma, const float* __restrict__ beta, int n) {
  const int node = (int)(((long long)blockIdx.x * blockDim.x + threadIdx.x) >> 5);
  const int g = threadIdx.x & 31;
  if (node >= n) return;
  float* p = A + (size_t)node * 128 + g * 4;
  float4 v  = *(float4*)p;
  const float4 bb = *(const float4*)(b + g * 4);
  v.x = fmaxf(v.x + bb.x, 0.0f);
  v.y = fmaxf(v.y + bb.y, 0.0f);
  v.z = fmaxf(v.z + bb.z, 0.0f);
  v.w = fmaxf(v.w + bb.w, 0.0f);
  const float mu  = 0.25f * (v.x + v.y + v.z + v.w);
  const float ms  = 0.25f * (v.x * v.x + v.y * v.y + v.z * v.z + v.w * v.w);
  const float inv = rsqrtf(fmaxf(ms - mu * mu, 0.0f) + 1e-5f);
  const float4 gm = *(const float4*)(gamma + g * 4);
  const float4 bt = *(const float4*)(beta + g * 4);
  v.x = (v.x - mu) * inv * gm.x + bt.x;
  v.y = (v.y - mu) * inv * gm.y + bt.y;
  v.z = (v.z - mu) * inv * gm.z + bt.z;
  v.w = (v.w - mu) * inv * gm.w + bt.w;
  *(float4*)p = v;
}

// ---------------------------------------------------------------------------
// bias + log_softmax over 40 classes: one wave per node, shuffle reductions
// ---------------------------------------------------------------------------
__global__ __launch_bounds__(256) void lsm40(
    const float* __restrict__ Ain, const float* __restrict__ b,
    float* __restrict__ out, int n) {
  const int node = (int)(((long long)blockIdx.x * blockDim.x + threadIdx.x) >> 5);
  const int lane = threadIdx.x & 31;
  if (node >= n) return;
  const float* row = Ain + (size_t)node * 40;
  const float v0 = row[lane] + b[lane];
  const float v1 = (lane < 8) ? (row[32 + lane] + b[32 + lane]) : -INFINITY;
  float m = fmaxf(v0, v1);
#pragma unroll
  for (int o = 16; o > 0; o >>= 1) m = fmaxf(m, __shfl_xor(m, o, 32));
  float s = expf(v0 - m) + ((lane < 8) ? expf(v1 - m) : 0.0f);
#pragma unroll
  for (int o = 16; o > 0; o >>= 1) s += __shfl_xor(s, o, 32);
  const float lse = m + logf(s);
  out[(size_t)node * 40 + lane] = v0 - lse;
  if (lane < 8) out[(size_t)node * 40 + 32 + lane] = v1 - lse;
}

// ---------------------------------------------------------------------------
extern "C" void kernel_launch(void* const* d_in, const int* in_sizes, int n_in,
                              void* d_out, int out_size, void* d_ws, size_t ws_size,
                              hipStream_t stream) {
  const int N = NNODES, E = NEDGES;
  const float* x     = (const float*)d_in[0];
  const int*   src   = (const int*)d_in[1];
  const int*   tgt   = (const int*)d_in[2];
  const float* w     = (const float*)d_in[3];
  const float* W0    = (const float*)d_in[4];
  const float* b0    = (const float*)d_in[5];
  const float* W1    = (const float*)d_in[6];
  const float* b1    = (const float*)d_in[7];
  const float* W2    = (const float*)d_in[8];
  const float* b2    = (const float*)d_in[9];
  const float* gamma = (const float*)d_in[10];
  const float* beta  = (const float*)d_in[11];
  float* out = (float*)d_out;

  float* H   = (float*)d_ws;                // N x 128
  float* A   = H + (size_t)N * 128;         // N x 128
  float* H2  = H;                            // N x 40 (reuses H)
  float* AG2 = H + (size_t)N * 64;          // N x 40 (disjoint from H2)

  const long long tot128 = (long long)N * 128;
  const long long tot40  = (long long)N * 40;
  const int gemm_blocks  = (N + 127) / 128;           // 8 waves x 16 rows per block
  const int edge_blocks  = (E + 7) / 8;               // 8 edges (waves) per block
  const int ew_blocks    = (int)((tot128 + 255) / 256);
  const int node_blocks  = (N * 32 + 255) / 256;      // wave-per-node kernels

  // ---- layer 0: relu(agg(x @ W0) + b0) ----
  gemm_wmma_f32<128><<<gemm_blocks, 256, 0, stream>>>(x, W0, H, N);
  zerof<<<ew_blocks, 256, 0, stream>>>(A, tot128);
  agg128<<<edge_blocks, 256, 0, stream>>>(H, src, tgt, w, A, E);
  bias_relu128<<<ew_blocks, 256, 0, stream>>>(A, b0, tot128);

  // ---- layer 1: groupnorm(relu(agg(h @ W1) + b1)) ----
  gemm_wmma_f32<128><<<gemm_blocks, 256, 0, stream>>>(A, W1, H, N);
  zerof<<<ew_blocks, 256, 0, stream>>>(A, tot128);
  agg128<<<edge_blocks, 256, 0, stream>>>(H, src, tgt, w, A, E);
  gn_bias_relu<<<node_blocks, 256, 0, stream>>>(A, b1, gamma, beta, N);

  // ---- layer 2: log_softmax(agg(h @ W2) + b2) ----
  gemm_wmma_f32<40><<<gemm_blocks, 256, 0, stream>>>(A, W2, H2, N);
  zerof<<<(int)((tot40 + 255) / 256), 256, 0, stream>>>(AG2, tot40);
  agg40<<<edge_blocks, 256, 0, stream>>>(H2, src, tgt, w, AG2, E);
  lsm40<<<node_blocks, 256, 0, stream>>>(AG2, b2, out, N);
}